// AttentionPooling_910533067558
// MI455X (gfx1250) — compile-verified
//
#include <hip/hip_runtime.h>
#include <hip/hip_bf16.h>
#include <math.h>

typedef float v2f __attribute__((ext_vector_type(2)));
typedef float v8f __attribute__((ext_vector_type(8)));

#define DIMD 512
#define ROWS 128           // rows of x per block (8 row-tiles of 16)
#define XS_STRIDE 516      // 516 % 64 = 4 -> conflict-free strided LDS reads

// ---------------- init: smax = -inf, gsum = 0 ----------------
__global__ void init_seg_kernel(float* __restrict__ smax, float* __restrict__ gsum, int ns) {
  int i = blockIdx.x * blockDim.x + threadIdx.x;
  if (i < ns) { smax[i] = -__builtin_inff(); gsum[i] = 0.0f; }
}

// ---------------- pass 1: gate = x @ Wg + bg (one wave per row) ----------------
__global__ void gate_gemv_kernel(const float* __restrict__ x, const float* __restrict__ Wg,
                                 const float* __restrict__ bg, float* __restrict__ gate, int n) {
  int wave = (blockIdx.x * blockDim.x + threadIdx.x) >> 5;
  int lane = threadIdx.x & 31;
  if (wave >= n) return;
  const float4* xr = (const float4*)(x + (size_t)wave * DIMD + lane * 16);
  const float4* wg = (const float4*)(Wg + lane * 16);
  float acc = 0.0f;
#pragma unroll
  for (int j = 0; j < 4; ++j) {
    float4 xv = xr[j]; float4 wv = wg[j];
    acc += xv.x * wv.x + xv.y * wv.y + xv.z * wv.z + xv.w * wv.w;
  }
  for (int off = 16; off; off >>= 1) acc += __shfl_down(acc, off, 32);
  if (lane == 0) gate[wave] = acc + bg[0];
}

// ---------------- pass 2: segment max (int-punned float atomic max) ----------------
__device__ inline void atomicMaxFloat(float* addr, float value) {
  if (value >= 0.0f) atomicMax((int*)addr, __float_as_int(value));
  else               atomicMin((unsigned int*)addr, __float_as_uint(value));
}
__global__ void seg_max_kernel(const float* __restrict__ gate, const int* __restrict__ idx,
                               float* __restrict__ smax, int n) {
  int i = blockIdx.x * blockDim.x + threadIdx.x;
  if (i < n) atomicMaxFloat(&smax[idx[i]], gate[i]);
}

// ---------------- pass 3: egate = exp(gate - smax[idx]); gsum += egate ----------------
__global__ void seg_expsum_kernel(const float* __restrict__ gate, const int* __restrict__ idx,
                                  const float* __restrict__ smax, float* __restrict__ egate,
                                  float* __restrict__ gsum, int n) {
  int i = blockIdx.x * blockDim.x + threadIdx.x;
  if (i >= n) return;
  int s = idx[i];
  float e = expf(gate[i] - smax[s]);
  egate[i] = e;
  atomicAdd(&gsum[s], e);
}

// ---------------- pass 4: out[idx[i]] += w_i * (x_i @ Wm + bm), WMMA fp32 ----------------
// 512 threads = 16 waves. Block owns 128 rows (8 row-tiles) staged in 264KB LDS.
// Each wave owns 2 column tiles; per K-step one B fragment feeds 8 WMMAs
// (8 row-tile accumulators), so Wm is read from L2 exactly once per 128 rows.
__global__ void __launch_bounds__(512, 1)
fused_gemm_scatter_kernel(const float* __restrict__ x, const int* __restrict__ idx,
                          const float* __restrict__ egate, const float* __restrict__ gsum,
                          const float* __restrict__ Wm, const float* __restrict__ bm,
                          float* __restrict__ out, int n) {
  extern __shared__ float smem[];
  float* xs   = smem;                        // ROWS * XS_STRIDE
  float* wrow = smem + ROWS * XS_STRIDE;     // ROWS gate weights
  int*   srow = (int*)(wrow + ROWS);         // ROWS segment ids

  const int r0 = blockIdx.x * ROWS;
  const int t  = threadIdx.x;

  // Stage 128x512 x-tile into LDS (float4, padded row stride).
  for (int i = t; i < ROWS * (DIMD / 4); i += blockDim.x) {
    int row = i >> 7;            // 128 float4 per row
    int c4  = i & 127;
    int rg  = r0 + row; if (rg >= n) rg = n - 1;
    float4 v = ((const float4*)(x + (size_t)rg * DIMD))[c4];
    *(float4*)(xs + row * XS_STRIDE + c4 * 4) = v;
  }
  if (t < ROWS) {
    int rg = r0 + t;
    if (rg < n) {
      int s = idx[rg];
      srow[t] = s;
      wrow[t] = egate[rg] / (gsum[s] + 1e-10f);
    } else { srow[t] = 0; wrow[t] = 0.0f; }
  }
  __syncthreads();

  const int wave = t >> 5;         // 0..15
  const int lane = t & 31;
  const int hi   = lane >> 4;      // lane half selects K pair (A/B) and M half (C)
  const int nl   = lane & 15;

  for (int tci = 0; tci < 2; ++tci) {
    const int ct   = wave + tci * 16;      // column tile 0..31
    const int ncol = ct * 16 + nl;
    const float* wmp = Wm + ncol;
    v8f c[8];
#pragma unroll
    for (int rt = 0; rt < 8; ++rt) c[rt] = (v8f){};

#pragma unroll 2
    for (int k0 = 0; k0 < DIMD; k0 += 4) {
      const int kk = k0 + hi * 2;
      v2f b;
      b.x = wmp[(size_t)kk * DIMD];
      b.y = wmp[(size_t)(kk + 1) * DIMD];
#pragma unroll
      for (int rt = 0; rt < 8; ++rt) {
        v2f a = *(const v2f*)(xs + (rt * 16 + nl) * XS_STRIDE + kk);
        c[rt] = __builtin_amdgcn_wmma_f32_16x16x4_f32(false, a, false, b,
                                                      (short)0, c[rt], false, false);
      }
    }

    const float bmc = bm[ncol];
#pragma unroll
    for (int rt = 0; rt < 8; ++rt) {
#pragma unroll
      for (int v = 0; v < 8; ++v) {
        int lrow = rt * 16 + v + hi * 8;   // C layout: VGPR v -> M = v (+8 for upper half)
        float val = (c[rt][v] + bmc) * wrow[lrow];
        atomicAdd(out + (size_t)srow[lrow] * DIMD + ncol, val);
      }
    }
  }
}

extern "C" void kernel_launch(void* const* d_in, const int* in_sizes, int n_in,
                              void* d_out, int out_size, void* d_ws, size_t ws_size,
                              hipStream_t stream) {
  const float* x     = (const float*)d_in[0];
  const int*   index = (const int*)d_in[1];   // per-harness convention: integer -> const int*
  const float* Wg    = (const float*)d_in[2];
  const float* bg    = (const float*)d_in[3];
  const float* Wm    = (const float*)d_in[4];
  const float* bm    = (const float*)d_in[5];
  float* out = (float*)d_out;

  const int n  = in_sizes[0] / DIMD;   // 400000
  const int ns = out_size / DIMD;      // 50000

  float* gate  = (float*)d_ws;
  float* egate = gate + n;
  float* smax  = egate + n;
  float* gsum  = smax + ns;

  hipMemsetAsync(d_out, 0, (size_t)out_size * sizeof(float), stream);
  init_seg_kernel<<<(ns + 255) / 256, 256, 0, stream>>>(smax, gsum, ns);
  gate_gemv_kernel<<<(n + 7) / 8, 256, 0, stream>>>(x, Wg, bg, gate, n);
  seg_max_kernel<<<(n + 255) / 256, 256, 0, stream>>>(gate, index, smax, n);
  seg_expsum_kernel<<<(n + 255) / 256, 256, 0, stream>>>(gate, index, smax, egate, gsum, n);

  size_t shmem = (size_t)(ROWS * XS_STRIDE + ROWS) * sizeof(float) + ROWS * sizeof(int);
  fused_gemm_scatter_kernel<<<(n + ROWS - 1) / ROWS, 512, shmem, stream>>>(
      x, index, egate, gsum, Wm, bm, out, n);
}